// TrajPreLocalAttnLong_63230508532522
// MI455X (gfx1250) — compile-verified
//
#include <hip/hip_runtime.h>
#include <math.h>

#define SEQ    24
#define LOCEMB 512
#define TIMEMB 32
#define INPD   544
#define HID    1024
#define G4     4096
#define NUMC   200001      // NUM_LOCS + 1
#define FCK    2048        // 2*HID
#define NEGV   -100000000000.0f

typedef float v2f __attribute__((ext_vector_type(2)));
typedef float v8f __attribute__((ext_vector_type(8)));

// ---------------------------------------------------------------------------
// K1: x[t] = concat(emb_loc[loc[t]], emb_tim[tim[t]]);  xg = x @ W_ih^T + b
// grid (16, 24) x 256
// ---------------------------------------------------------------------------
__global__ void k_embed_xgates(const int* __restrict__ loc, const int* __restrict__ tim,
                               const float* __restrict__ emb_loc, const float* __restrict__ emb_tim,
                               const float* __restrict__ W_ih,
                               const float* __restrict__ b_ih, const float* __restrict__ b_hh,
                               float* __restrict__ xg) {
    __shared__ float x[INPD];
    const int t   = blockIdx.y;
    const int tid = threadIdx.x;
    const int l   = loc[t];
    const int tm  = tim[t];
    for (int k = tid; k < INPD; k += blockDim.x)
        x[k] = (k < LOCEMB) ? emb_loc[(size_t)l * LOCEMB + k]
                            : emb_tim[tm * TIMEMB + (k - LOCEMB)];
    __syncthreads();
    const int j = blockIdx.x * blockDim.x + tid;
    if (j < G4) {
        const float* w = W_ih + (size_t)j * INPD;
        float acc = b_ih[j] + b_hh[j];
        for (int k = 0; k < INPD; ++k) acc += x[k] * w[k];
        xg[t * G4 + j] = acc;
    }
}

// ---------------------------------------------------------------------------
// K2: one LSTM step. W_hh (16.8 MB) is L2-resident after step 0.
// grid 16 x 256: thread = (gate, j); tid<64 does the c/h update.
// ---------------------------------------------------------------------------
__global__ void k_lstm_step(const float* __restrict__ xg, const float* __restrict__ W_hh,
                            float* __restrict__ hs, float* __restrict__ c, int t) {
    __shared__ __align__(16) float hsh[HID];
    __shared__ float g4[4][64];
    const int tid = threadIdx.x;
    for (int k = tid; k < HID; k += 256)
        hsh[k] = (t == 0) ? 0.0f : hs[(t - 1) * HID + k];
    __syncthreads();
    const int gate = tid >> 6, jj = tid & 63;
    const int j   = blockIdx.x * 64 + jj;
    const int row = gate * HID + j;
    float acc = xg[t * G4 + row];
    const float4* w  = (const float4*)(W_hh + (size_t)row * HID);
    const float4* hv = (const float4*)hsh;
    for (int k = 0; k < HID / 4; ++k) {
        float4 wv = w[k], xv = hv[k];
        acc += wv.x * xv.x + wv.y * xv.y + wv.z * xv.z + wv.w * xv.w;
    }
    g4[gate][jj] = acc;
    __syncthreads();
    if (tid < 64) {
        const int jg = blockIdx.x * 64 + tid;
        const float si = 1.f / (1.f + __expf(-g4[0][tid]));
        const float sf = 1.f / (1.f + __expf(-g4[1][tid]));
        const float gg = tanhf(g4[2][tid]);
        const float so = 1.f / (1.f + __expf(-g4[3][tid]));
        const float cp = (t == 0) ? 0.f : c[jg];
        const float cn = sf * cp + si * gg;
        c[jg] = cn;
        hs[t * HID + jg] = so * tanhf(cn);
    }
}

// ---------------------------------------------------------------------------
// K3: causal self-attention (24x24) + build padded out matrix (32 x 2048),
// rows 24..31 zero.  Single block.
// ---------------------------------------------------------------------------
__global__ void k_attn_out(const float* __restrict__ hs, float* __restrict__ outbuf) {
    __shared__ float E[SEQ][SEQ];
    const int tid = threadIdx.x;
    for (int p = tid; p < SEQ * SEQ; p += blockDim.x) {
        const int i = p / SEQ, j = p % SEQ;
        float e = NEGV;
        if (j <= i) {
            const float* a = hs + i * HID;
            const float* b = hs + j * HID;
            e = 0.f;
            for (int k = 0; k < HID; ++k) e += a[k] * b[k];
        }
        E[i][j] = e;
    }
    __syncthreads();
    if (tid < SEQ) {
        const int i = tid;
        float mx = -INFINITY;
        for (int j = 0; j <= i; ++j) mx = fmaxf(mx, E[i][j]);
        float s = 0.f;
        for (int j = 0; j <= i; ++j) s += __expf(E[i][j] - mx);
        const float inv = 1.f / s;
        for (int j = 0; j < SEQ; ++j)
            E[i][j] = (j <= i) ? __expf(E[i][j] - mx) * inv : 0.f;
    }
    __syncthreads();
    for (int idx = tid; idx < 32 * FCK; idx += blockDim.x) {
        const int m = idx >> 11, k = idx & (FCK - 1);
        float v = 0.f;
        if (m < SEQ) {
            if (k < HID) v = hs[m * HID + k];
            else {
                const int kk = k - HID;
                for (int j = 0; j <= m; ++j) v += E[m][j] * hs[j * HID + kk];
            }
        }
        outbuf[idx] = v;
    }
}

// ---------------------------------------------------------------------------
// K4a: y = out @ fc_W^T + fc_b   via  V_WMMA_F32_16X16X4_F32.
// Each wave owns TWO 16-column tiles of fc_W (32 rows) so the cached A loads
// are amortized: per K-step = 2 NT stream loads + 2 cached loads -> 4 WMMAs.
// fc_W is streamed from HBM exactly once with TH=NT (no L2 pollution);
// A (out, 32x2048 padded) stays L2/L0-resident.
//   A lane layout (16x4 f32): M = lane&15, K-half = lane>>4, v2f = K pair
//   B lane layout (4x16 f32): same K-half split, N = lane&15
//   C/D layout: VGPR v -> M = v + 8*(lane>>4), N = lane&15
// ---------------------------------------------------------------------------
__global__ void k_fc_wmma(const float* __restrict__ outbuf,
                          const float* __restrict__ fcW,
                          const float* __restrict__ fcb,
                          float* __restrict__ y) {
    const int tid  = threadIdx.x;
    const int wave = tid >> 5, lane = tid & 31;
    const long n0  = ((long)blockIdx.x * 8 + wave) * 32;   // 32 fc rows per wave
    if (n0 >= NUMC) return;
    const int  r   = lane & 15, kh = lane >> 4;
    const long nA  = n0 + r;
    const long nB  = n0 + 16 + r;
    const long ncA = (nA < NUMC) ? nA : (NUMC - 1);
    const long ncB = (nB < NUMC) ? nB : (NUMC - 1);

    const float* bpA = fcW    + (size_t)ncA * FCK + 2 * kh;  // B tile 0 stream
    const float* bpB = fcW    + (size_t)ncB * FCK + 2 * kh;  // B tile 1 stream
    const float* a0  = outbuf + (size_t)r   * FCK + 2 * kh;  // A rows 0..15
    const float* a1  = a0 + (size_t)16 * FCK;                // A rows 16..31

    v8f acc00 = {}, acc01 = {}, acc10 = {}, acc11 = {};
#pragma unroll 4
    for (int k = 0; k < FCK; k += 4) {
        v2f bA = __builtin_nontemporal_load((const v2f*)(bpA + k));
        v2f bB = __builtin_nontemporal_load((const v2f*)(bpB + k));
        v2f x0 = *(const v2f*)(a0 + k);
        v2f x1 = *(const v2f*)(a1 + k);
        acc00 = __builtin_amdgcn_wmma_f32_16x16x4_f32(false, x0, false, bA,
                                                      (short)0, acc00, false, false);
        acc10 = __builtin_amdgcn_wmma_f32_16x16x4_f32(false, x1, false, bA,
                                                      (short)0, acc10, false, false);
        acc01 = __builtin_amdgcn_wmma_f32_16x16x4_f32(false, x0, false, bB,
                                                      (short)0, acc01, false, false);
        acc11 = __builtin_amdgcn_wmma_f32_16x16x4_f32(false, x1, false, bB,
                                                      (short)0, acc11, false, false);
    }

    if (nA < NUMC) {
        const float bias = fcb[nA];
#pragma unroll
        for (int v = 0; v < 8; ++v) {
            const int m0 = v + 8 * kh;                 // rows 0..15
            y[(size_t)m0 * NUMC + nA] = acc00[v] + bias;
            const int m1 = 16 + v + 8 * kh;            // rows 16..23 valid
            if (m1 < SEQ) y[(size_t)m1 * NUMC + nA] = acc10[v] + bias;
        }
    }
    if (nB < NUMC) {
        const float bias = fcb[nB];
#pragma unroll
        for (int v = 0; v < 8; ++v) {
            const int m0 = v + 8 * kh;
            y[(size_t)m0 * NUMC + nB] = acc01[v] + bias;
            const int m1 = 16 + v + 8 * kh;
            if (m1 < SEQ) y[(size_t)m1 * NUMC + nB] = acc11[v] + bias;
        }
    }
}

// ---------------------------------------------------------------------------
// K4b/c/d: online (max, sum-exp) reduction per row, then normalize in place.
// ---------------------------------------------------------------------------
__device__ __forceinline__ void merge_ms(float& m1, float& s1, float m2, float s2) {
    if (m2 > m1) { s1 = s1 * __expf(m1 - m2) + s2; m1 = m2; }
    else         { s1 = s1 + s2 * __expf(m2 - m1); }
}

__global__ void k_row_partial(const float* __restrict__ y, float* __restrict__ partials) {
    const int m = blockIdx.y, chunk = blockIdx.x, tid = threadIdx.x;
    const int CH = (NUMC + 31) / 32;
    const int s  = chunk * CH;
    const int e  = (s + CH < NUMC) ? (s + CH) : NUMC;
    const float* row = y + (size_t)m * NUMC;
    float lm = -INFINITY, ls = 0.f;
    for (int i = s + tid; i < e; i += 256) {
        const float v = row[i];
        if (v > lm) { ls = ls * __expf(lm - v) + 1.f; lm = v; }
        else          ls += __expf(v - lm);
    }
    __shared__ float sm[256], ss[256];
    sm[tid] = lm; ss[tid] = ls;
    __syncthreads();
    for (int st = 128; st > 0; st >>= 1) {
        if (tid < st) {
            float a = sm[tid], b = ss[tid];
            merge_ms(a, b, sm[tid + st], ss[tid + st]);
            sm[tid] = a; ss[tid] = b;
        }
        __syncthreads();
    }
    if (tid == 0) {
        partials[(m * 32 + chunk) * 2]     = sm[0];
        partials[(m * 32 + chunk) * 2 + 1] = ss[0];
    }
}

__global__ void k_row_merge(const float* __restrict__ partials, float* __restrict__ stats) {
    const int m = blockIdx.x, tid = threadIdx.x;   // 32 threads
    __shared__ float sm[32], ss[32];
    sm[tid] = partials[(m * 32 + tid) * 2];
    ss[tid] = partials[(m * 32 + tid) * 2 + 1];
    __syncthreads();
    for (int st = 16; st > 0; st >>= 1) {
        if (tid < st) {
            float a = sm[tid], b = ss[tid];
            merge_ms(a, b, sm[tid + st], ss[tid + st]);
            sm[tid] = a; ss[tid] = b;
        }
        __syncthreads();
    }
    if (tid == 0) { stats[m * 2] = sm[0]; stats[m * 2 + 1] = __logf(ss[0]); }
}

__global__ void k_normalize(float* __restrict__ y, const float* __restrict__ stats) {
    const long i = (long)blockIdx.x * blockDim.x + threadIdx.x;
    if (i >= (long)SEQ * NUMC) return;
    const int m = (int)(i / NUMC);
    y[i] -= stats[m * 2] + stats[m * 2 + 1];
}

// ---------------------------------------------------------------------------
extern "C" void kernel_launch(void* const* d_in, const int* in_sizes, int n_in,
                              void* d_out, int out_size, void* d_ws, size_t ws_size,
                              hipStream_t stream) {
    (void)in_sizes; (void)n_in; (void)out_size; (void)ws_size;
    const int*   loc     = (const int*)d_in[0];
    const int*   tim     = (const int*)d_in[1];
    // d_in[2] = target_len (== SEQ)
    const float* emb_loc = (const float*)d_in[3];
    const float* emb_tim = (const float*)d_in[4];
    const float* W_ih    = (const float*)d_in[5];
    const float* W_hh    = (const float*)d_in[6];
    const float* b_ih    = (const float*)d_in[7];
    const float* b_hh    = (const float*)d_in[8];
    const float* fcW     = (const float*)d_in[9];
    const float* fcb     = (const float*)d_in[10];
    float* y  = (float*)d_out;
    float* ws = (float*)d_ws;

    float* xg       = ws;                       // 24*4096
    float* hs       = xg + SEQ * G4;            // 24*1024
    float* c        = hs + SEQ * HID;           // 1024
    float* outbuf   = c + HID;                  // 32*2048 (16B-aligned offset)
    float* partials = outbuf + 32 * FCK;        // 24*32*2
    float* stats    = partials + SEQ * 32 * 2;  // 24*2

    k_embed_xgates<<<dim3(16, SEQ), 256, 0, stream>>>(loc, tim, emb_loc, emb_tim,
                                                      W_ih, b_ih, b_hh, xg);
    for (int t = 0; t < SEQ; ++t)
        k_lstm_step<<<16, 256, 0, stream>>>(xg, W_hh, hs, c, t);
    k_attn_out<<<1, 256, 0, stream>>>(hs, outbuf);

    const int ntiles32 = (NUMC + 31) / 32;       // 6251 wave-tiles
    const int nblocks  = (ntiles32 + 7) / 8;     // 782 blocks x 8 waves
    k_fc_wmma<<<nblocks, 256, 0, stream>>>(outbuf, fcW, fcb, y);

    k_row_partial<<<dim3(32, SEQ), 256, 0, stream>>>(y, partials);
    k_row_merge<<<SEQ, 32, 0, stream>>>(partials, stats);
    const long tot = (long)SEQ * NUMC;
    k_normalize<<<(int)((tot + 255) / 256), 256, 0, stream>>>(y, stats);
}